// GNNRecommendationModel_18485539242072
// MI455X (gfx1250) — compile-verified
//
#include <hip/hip_runtime.h>

// GCN (2-layer) on MI455X / gfx1250, wave32, f32 WMMA path.

typedef __attribute__((ext_vector_type(2))) float v2f;
typedef __attribute__((ext_vector_type(8))) float v8f;

#define NN    100000   // total nodes
#define DD    64       // embedding dim
#define SPLIT 50000    // user/item boundary (divisible by 16 -> tiles never straddle)

// ---------------------------------------------------------------- degree / norm
__global__ void k_deg_init(float* __restrict__ deg, int n) {
  int i = blockIdx.x * blockDim.x + threadIdx.x;
  if (i < n) deg[i] = 1.0f;  // self-loop contributes 1 to every node's degree
}

__global__ void k_deg_edges(const int* __restrict__ col, float* __restrict__ deg, int E) {
  int e = blockIdx.x * blockDim.x + threadIdx.x;
  if (e < E) atomicAdd(&deg[col[e]], 1.0f);
}

__global__ void k_dinv(float* __restrict__ deg, int n) {
  int i = blockIdx.x * blockDim.x + threadIdx.x;
  if (i < n) deg[i] = rsqrtf(deg[i]);  // deg >= 1 always (self-loop)
}

// ---------------------------------------------------------------- WMMA GEMM
// out[nrows x 64] = A[nrows x 64] @ W[64 x 64], A = concat(A0, A1) split at SPLIT.
// One wave -> one 16x64 output tile using V_WMMA_F32_16X16X4_F32.
// A frag (16x4 f32, 2 VGPR): lane L: M = L&15; K = {0,1} for L<16, {2,3} for L>=16.
// B frag (4x16 f32, 2 VGPR): lane L: N = L&15; K pair mirrors A.
// C/D (16x16 f32, 8 VGPR):   VGPR r: lanes 0-15 -> M=r, lanes 16-31 -> M=r+8; N = L&15.
//
// W is staged in LDS K-pair interleaved:  sW[kp*64 + n] = {W[2kp][n], W[2kp+1][n]}
// so each B fragment is one contiguous ds_load_b64 per lane (no register shuffles).
template <int RELU>
__global__ void k_gemm(const float* __restrict__ A0, const float* __restrict__ A1,
                       const float* __restrict__ W, float* __restrict__ out, int nrows) {
  __shared__ float2 sW[(DD / 2) * DD];  // 16 KB, pair-interleaved
  for (int i = threadIdx.x; i < (DD / 2) * DD; i += blockDim.x) {
    int kp = i >> 6;
    int n  = i & 63;
    float2 p;
    p.x = W[(2 * kp)     * DD + n];
    p.y = W[(2 * kp + 1) * DD + n];
    sW[i] = p;
  }
  __syncthreads();

  const int lane = threadIdx.x & 31;
  const int wave = threadIdx.x >> 5;
  const int m0 = (blockIdx.x * 8 + wave) * 16;  // wave-uniform -> EXEC stays all-1s
  if (m0 >= nrows) return;

  const int mr   = lane & 15;
  const int half = lane >> 4;   // 0 -> K pair {0,1}, 1 -> K pair {2,3}
  const int row  = m0 + mr;
  const float* __restrict__ arow =
      (row < SPLIT) ? (A0 + (size_t)row * DD) : (A1 + (size_t)(row - SPLIT) * DD);

  v8f acc0 = {}; v8f acc1 = {}; v8f acc2 = {}; v8f acc3 = {};
  const int nc = lane & 15;

#pragma unroll
  for (int k0 = 0; k0 < DD; k0 += 4) {
    const int ka = k0 + half * 2;
    float ax = arow[ka];
    float ay = arow[ka + 1];
    if (RELU) { ax = fmaxf(ax, 0.0f); ay = fmaxf(ay, 0.0f); }
    v2f a; a.x = ax; a.y = ay;

    // B fragments: kp = ka/2 selects the pair row; lanes>=16 naturally get kp+1
    const v2f* __restrict__ wp = (const v2f*)&sW[(ka >> 1) * DD + nc];
    v2f b0 = wp[0];
    v2f b1 = wp[16];
    v2f b2 = wp[32];
    v2f b3 = wp[48];

    acc0 = __builtin_amdgcn_wmma_f32_16x16x4_f32(false, a, false, b0, (short)0, acc0, false, false);
    acc1 = __builtin_amdgcn_wmma_f32_16x16x4_f32(false, a, false, b1, (short)0, acc1, false, false);
    acc2 = __builtin_amdgcn_wmma_f32_16x16x4_f32(false, a, false, b2, (short)0, acc2, false, false);
    acc3 = __builtin_amdgcn_wmma_f32_16x16x4_f32(false, a, false, b3, (short)0, acc3, false, false);
  }

  const int rbase = m0 + half * 8;
#pragma unroll
  for (int i = 0; i < 8; i++) {
    float* orow = out + (size_t)(rbase + i) * DD + nc;
    orow[0]  = acc0[i];
    orow[16] = acc1[i];
    orow[32] = acc2[i];
    orow[48] = acc3[i];
  }
}

// ---------------------------------------------------------------- self-loop + bias init
// out[v,:] = dinv[v]^2 * xw[v,:] + b   (initializes accumulator -> gives launch ordering)
__global__ void k_self_init(const float* __restrict__ xw, const float* __restrict__ dinv,
                            const float* __restrict__ bias, float* __restrict__ out, int n) {
  long idx = (long)blockIdx.x * blockDim.x + threadIdx.x;  // over n*16 float4s
  if (idx >= (long)n * 16) return;
  int v  = (int)(idx >> 4);
  int d4 = (int)(idx & 15);
  float s = dinv[v];
  s = s * s;
  float4 x = ((const float4*)xw)[idx];
  float4 b = ((const float4*)bias)[d4];
  float4 o;
  o.x = s * x.x + b.x; o.y = s * x.y + b.y;
  o.z = s * x.z + b.z; o.w = s * x.w + b.w;
  ((float4*)out)[idx] = o;
}

// ---------------------------------------------------------------- edge gather/scatter
// One thread per (edge, dim): out[col,d] += dinv[row]*dinv[col] * xw[row,d]
// xw (25.6 MB) and out (25.6 MB) are L2-resident (192 MB L2) -> on-chip atomics.
__global__ void k_edge_agg(const int* __restrict__ row, const int* __restrict__ col,
                           const float* __restrict__ dinv, const float* __restrict__ xw,
                           float* __restrict__ out, int E) {
  long idx = (long)blockIdx.x * blockDim.x + threadIdx.x;
  long e = idx >> 6;
  if (e >= E) return;
  int d = (int)(idx & 63);
  int r = row[e];
  int c = col[e];
  float nrm = dinv[r] * dinv[c];
  atomicAdd(out + (size_t)c * DD + d, nrm * xw[(size_t)r * DD + d]);
}

// ---------------------------------------------------------------- launch
extern "C" void kernel_launch(void* const* d_in, const int* in_sizes, int n_in,
                              void* d_out, int out_size, void* d_ws, size_t ws_size,
                              hipStream_t stream) {
  const int* edge = (const int*)d_in[0];
  const int  E    = in_sizes[0] / 2;
  const int* erow = edge;       // edge_index[0]
  const int* ecol = edge + E;   // edge_index[1]
  const float* user_emb = (const float*)d_in[1];
  const float* item_emb = (const float*)d_in[2];
  const float* W1 = (const float*)d_in[3];
  const float* b1 = (const float*)d_in[4];
  const float* W2 = (const float*)d_in[5];
  const float* b2 = (const float*)d_in[6];
  float* out = (float*)d_out;

  // workspace layout (dinv padded to 256 B so xw/h1 stay 16 B aligned)
  float* dinv = (float*)d_ws;                 // 100352 floats
  float* xw   = dinv + 100352;                // NN*DD
  float* h1   = xw + (size_t)NN * DD;         // NN*DD

  const int T = 256;
  const long n16 = (long)NN * 16;             // float4 elements per matrix
  const long ne  = (long)E * 64;              // (edge, dim) pairs
  const int gemm_blocks = (NN / 16 + 7) / 8;  // 8 waves (tiles) per block

  // normalization: deg = 1 + in-degree; dinv = rsqrt(deg)
  k_deg_init <<<(NN + T - 1) / T, T, 0, stream>>>(dinv, NN);
  k_deg_edges<<<(E  + T - 1) / T, T, 0, stream>>>(ecol, dinv, E);
  k_dinv     <<<(NN + T - 1) / T, T, 0, stream>>>(dinv, NN);

  // layer 1: xw = X @ W1 ; h1 = dinv^2*xw + b1 (self-loops) ; += edge messages
  k_gemm<0>  <<<gemm_blocks, T, 0, stream>>>(user_emb, item_emb, W1, xw, NN);
  k_self_init<<<(n16 + T - 1) / T, T, 0, stream>>>(xw, dinv, b1, h1, NN);
  k_edge_agg <<<(ne  + T - 1) / T, T, 0, stream>>>(erow, ecol, dinv, xw, h1, E);

  // layer 2: xw = relu(h1) @ W2 ; out = dinv^2*xw + b2 ; += edge messages
  k_gemm<1>  <<<gemm_blocks, T, 0, stream>>>(h1, h1 + (size_t)SPLIT * DD, W2, xw, NN);
  k_self_init<<<(n16 + T - 1) / T, T, 0, stream>>>(xw, dinv, b2, out, NN);
  k_edge_agg <<<(ne  + T - 1) / T, T, 0, stream>>>(erow, ecol, dinv, xw, out, E);
}